// GCN_7756710936726
// MI455X (gfx1250) — compile-verified
//
#include <hip/hip_runtime.h>
#include <hip/hip_bf16.h>

typedef __attribute__((ext_vector_type(2))) float v2f;
typedef __attribute__((ext_vector_type(8))) float v8f;

// ---------------------------------------------------------------------------
// Dense GEMM C[M,N] = A[M,K] * B[K,N] using V_WMMA_F32_16X16X4_F32 (wave32).
//
// One wave owns one 16-wide column tile and keeps its entire B tile (K x 16,
// = K/4 float2 per lane = 64 VGPRs at K=128) resident in registers, reusing
// it across R consecutive 16-row tiles. K is a compile-time constant so the
// inner loop fully unrolls into a straight line of global_load_b64 + v_wmma.
//
// Out-of-range columns (layer 2: N=40 inside 48) are handled WITHOUT EXEC
// masking: the B column index is clamped in-bounds; garbage columns only
// affect D columns >= N, which are never stored. EXEC stays all-1s at every
// WMMA as the ISA requires.
//
// VGPR layouts (ISA 7.12.2, wave32):
//   A 16x4:  lane l -> A[l&15][k0 + 2*(l>=16) + {0,1}]   (contiguous float2)
//   B 4x16:  lane l -> B[k0 + 2*(l>=16) + {0,1}][col0 + (l&15)]
//   C/D:     elem v, lane l -> C[row0 + v + 8*(l>=16)][col0 + (l&15)]
// ---------------------------------------------------------------------------
template <int K, int R>
__global__ void __launch_bounds__(128)
gcn_gemm_wmma_f32(const float* __restrict__ A, const float* __restrict__ B,
                  float* __restrict__ C, int N)
{
    const int lane = threadIdx.x & 31;
    const int wave = threadIdx.x >> 5;
    const int col0 = wave << 4;
    const int m16  = lane & 15;   // M index for A-loads, N index for B/C
    const int hi   = lane >> 4;   // 0: lanes 0-15, 1: lanes 16-31
    const int n    = col0 + m16;
    const int nc   = (n < N) ? n : (N - 1);   // clamped, always in-bounds

    // Stage this wave's B tile (K x 16) in registers.
    v2f breg[K / 4];
    {
        const float* __restrict__ bp = B + (size_t)(hi << 1) * N + nc;
        #pragma unroll
        for (int i = 0; i < K / 4; ++i) {
            breg[i].x = bp[0];
            breg[i].y = bp[N];
            bp += (size_t)4 * N;
        }
    }

    #pragma unroll 1
    for (int r = 0; r < R; ++r) {
        const int row0 = (blockIdx.x * R + r) << 4;
        const float* __restrict__ arow =
            A + (size_t)(row0 + m16) * K + (hi << 1);

        v8f acc = {};
        #pragma unroll
        for (int i = 0; i < K / 4; ++i) {
            v2f a = *reinterpret_cast<const v2f*>(arow);
            arow += 4;
            // D = A*B + C : (neg_a, A, neg_b, B, c_mod, C, reuse_a, reuse_b)
            acc = __builtin_amdgcn_wmma_f32_16x16x4_f32(
                false, a, false, breg[i], (short)0, acc, false, false);
        }

        if (n < N) {
            #pragma unroll
            for (int v = 0; v < 8; ++v)
                C[(size_t)(row0 + v + (hi << 3)) * N + n] = acc[v];
        }
    }
}

// ---------------------------------------------------------------------------
// Elementwise / graph kernels (F is compile-time: 64 -> shifts, 40 -> const)
// ---------------------------------------------------------------------------
__global__ void __launch_bounds__(256)
k_zero_f32(float* __restrict__ p, int n)
{
    int i = blockIdx.x * blockDim.x + threadIdx.x;
    if (i < n) p[i] = 0.0f;
}

__global__ void __launch_bounds__(256)
k_count_deg(const int* __restrict__ dst, float* __restrict__ deg, int E)
{
    int i = blockIdx.x * blockDim.x + threadIdx.x;
    if (i < E) atomicAdd(&deg[dst[i]], 1.0f);
}

__global__ void __launch_bounds__(256)
k_make_dinv(const float* __restrict__ deg, float* __restrict__ dinv, int n)
{
    int i = blockIdx.x * blockDim.x + threadIdx.x;
    if (i < n) dinv[i] = __frsqrt_rn(deg[i] + 1.0f);  // +1 = self loop; deg>0 always
}

// agg[i,f] = h[i,f] * dinv[i]^2   (self-loop term; fully initializes agg)
template <int F>
__global__ void __launch_bounds__(256)
k_init_agg(const float* __restrict__ h, const float* __restrict__ dinv,
           float* __restrict__ agg, int n)
{
    int gid = blockIdx.x * blockDim.x + threadIdx.x;
    if (gid < n * F) {
        int i = gid / F;
        float dv = dinv[i];
        agg[gid] = h[gid] * dv * dv;
    }
}

// agg[dst,f] += h[src,f] * dinv[src] * dinv[dst]   (thread per edge*feature)
template <int F>
__global__ void __launch_bounds__(256)
k_scatter_edges(const int* __restrict__ src, const int* __restrict__ dst,
                const float* __restrict__ h, const float* __restrict__ dinv,
                float* __restrict__ agg, int E)
{
    long long gid = (long long)blockIdx.x * blockDim.x + threadIdx.x;
    if (gid >= (long long)E * F) return;
    int e = (int)(gid / F);
    int f = (int)(gid - (long long)e * F);
    int s = src[e];
    int d = dst[e];
    float v = h[(size_t)s * F + f] * dinv[s] * dinv[d];
    atomicAdd(&agg[(size_t)d * F + f], v);
}

// out[i,f] = relu(agg[i,f] + bias[f])
template <int F>
__global__ void __launch_bounds__(256)
k_bias_relu(const float* __restrict__ agg, const float* __restrict__ bias,
            float* __restrict__ out, int n)
{
    int gid = blockIdx.x * blockDim.x + threadIdx.x;
    if (gid < n * F) {
        int f = gid % F;
        float v = agg[gid] + bias[f];
        out[gid] = v > 0.0f ? v : 0.0f;
    }
}

// ---------------------------------------------------------------------------
// Launch
// ---------------------------------------------------------------------------
extern "C" void kernel_launch(void* const* d_in, const int* in_sizes, int n_in,
                              void* d_out, int out_size, void* d_ws, size_t ws_size,
                              hipStream_t stream)
{
    (void)n_in; (void)out_size; (void)ws_size;

    constexpr int FIN = 128, HID = 64, NCLS = 40;
    const int N = in_sizes[0] / FIN;     // 50000 (divisible by 16 and by 80)
    const int E = in_sizes[1] / 2;       // 800000

    const float* x   = (const float*)d_in[0];
    const int*   ei  = (const int*)d_in[1];
    const int*   src = ei;               // edge_index[0]
    const int*   dst = ei + E;           // edge_index[1]
    const float* W1  = (const float*)d_in[2];
    const float* b1  = (const float*)d_in[3];
    const float* W2  = (const float*)d_in[4];
    const float* b2  = (const float*)d_in[5];
    float*       out = (float*)d_out;

    // Workspace (floats), aliased across phases:
    //   h1 = bufA, agg1 = bufB, h1act = bufC, g = bufB (agg1 dead), agg2 = bufA (h1 dead)
    const size_t Np = (size_t)((N + 255) & ~255);
    float* deg   = (float*)d_ws;
    float* dinv  = deg + Np;
    float* bufA  = dinv + Np;
    float* bufB  = bufA + (size_t)N * HID;
    float* bufC  = bufB + (size_t)N * HID;

    float* h1    = bufA;   // x @ W1             [N,64]
    float* agg1  = bufB;   // aggregated layer 1 [N,64]
    float* h1act = bufC;   // relu(agg1 + b1)    [N,64]
    float* g     = bufB;   // h1act @ W2         [N,40]
    float* agg2  = bufA;   // aggregated layer 2 [N,40]

    constexpr int B256 = 256;
    constexpr int R    = 5;               // row tiles per wave (3125 = 625 * 5)
    auto blocks = [](long long work, int bs) { return (unsigned)((work + bs - 1) / bs); };
    const unsigned gemmGrid = (unsigned)(N / 16 / R);   // 625

    // --- normalization: deg -> dinv ---
    k_zero_f32 <<<blocks(N, B256), B256, 0, stream>>>(deg, N);
    k_count_deg<<<blocks(E, B256), B256, 0, stream>>>(dst, deg, E);
    k_make_dinv<<<blocks(N, B256), B256, 0, stream>>>(deg, dinv, N);

    // --- layer 1: h1 = x @ W1 (WMMA f32), aggregate, bias+relu ---
    gcn_gemm_wmma_f32<FIN, R><<<gemmGrid, 32 * (HID / 16), 0, stream>>>(x, W1, h1, HID);
    k_init_agg<HID>     <<<blocks((long long)N * HID, B256), B256, 0, stream>>>(h1, dinv, agg1, N);
    k_scatter_edges<HID><<<blocks((long long)E * HID, B256), B256, 0, stream>>>(src, dst, h1, dinv, agg1, E);
    k_bias_relu<HID>    <<<blocks((long long)N * HID, B256), B256, 0, stream>>>(agg1, b1, h1act, N);

    // --- layer 2: g = h1act @ W2 (WMMA f32, 3 col tiles masked to 40), aggregate, bias+relu ---
    constexpr int colTiles2 = (NCLS + 15) / 16;  // 3
    gcn_gemm_wmma_f32<HID, R><<<gemmGrid, 32 * colTiles2, 0, stream>>>(h1act, W2, g, NCLS);
    k_init_agg<NCLS>     <<<blocks((long long)N * NCLS, B256), B256, 0, stream>>>(g, dinv, agg2, N);
    k_scatter_edges<NCLS><<<blocks((long long)E * NCLS, B256), B256, 0, stream>>>(src, dst, g, dinv, agg2, E);
    k_bias_relu<NCLS>    <<<blocks((long long)N * NCLS, B256), B256, 0, stream>>>(agg2, b2, out, N);
}